// DetectionNetwork_54786602828357
// MI455X (gfx1250) — compile-verified
//
#include <hip/hip_runtime.h>
#include <stdint.h>

// YOLOv3 decode for MI455X (gfx1250).
// Bandwidth-bound transpose+elementwise: use the Tensor Data Mover to DMA
// an 85(ch) x 128(spatial) tile (row stride = H*W) from global into LDS with
// 1-DWORD row padding, then emit fully coalesced 85-channel-minor output.

typedef __attribute__((ext_vector_type(4))) unsigned int uint4v;
typedef __attribute__((ext_vector_type(8))) int         int8v;
typedef __attribute__((ext_vector_type(4))) int         int4v;

#define TILE_S    128          // spatial positions per tile
#define NCH       85           // 5 + 80 classes
#define ROWSTRIDE (TILE_S + 1) // padded LDS row stride (DWORDs) for full tiles

__launch_bounds__(256)
__global__ void yolo_decode_tdm(const float* __restrict__ in,
                                float* __restrict__ out,
                                int W, int HW,
                                float strideF,
                                int scaleRowOff,
                                float a0w, float a0h,
                                float a1w, float a1h,
                                float a2w, float a2h)
{
    __shared__ float lds[NCH * ROWSTRIDE];   // 85*129*4 = 43,860 B

    const int chunk = blockIdx.x;
    const int a     = blockIdx.y;
    const int b     = blockIdx.z;
    const int s0    = chunk * TILE_S;
    const int rem   = min(TILE_S, HW - s0);
    const int rowStride = (rem == TILE_S) ? ROWSTRIDE : rem;

    // start of channel-0 row for this (b, anchor) at spatial offset s0
    const int inOff = (b * 255 + a * NCH) * HW + s0;

    // ---- TDM: DMA the 85 x rem tile (global row stride HW) into LDS ----
    if (threadIdx.x < 32u) {   // one wave issues the tensor DMA
        uint64_t gaddr = (uint64_t)(uintptr_t)(in + inOff);
        uint32_t laddr = (uint32_t)(uintptr_t)(&lds[0]);  // flat low 32 = LDS offset

        uint4v g0;
        g0.x = 1u;                                        // count=1 valid descriptor
        g0.y = laddr;                                     // lds_addr (bytes)
        g0.z = (uint32_t)(gaddr & 0xFFFFFFFFu);           // global_addr[31:0]
        g0.w = (uint32_t)((gaddr >> 32) & 0x01FFFFFFu)    // global_addr[56:32]
             | (2u << 30);                                // type = 2 ("image")

        const uint32_t padEn = (rem == TILE_S) ? 1u : 0u;
        int8v g1;
        // data_size=2 (4B), pad_enable, pad_interval code 6 (128 DW), pad_amount code 0 (1 DW)
        g1[0] = (int)((2u << 16) | (padEn << 20) | (6u << 22));
        g1[1] = (int)(((uint32_t)rem & 0xFFFFu) << 16);   // tensor_dim0[15:0]
        g1[2] = (int)((((uint32_t)rem >> 16) & 0xFFFFu)   // tensor_dim0[31:16]
                     | ((uint32_t)NCH << 16));            // tensor_dim1[15:0] = 85
        g1[3] = (int)(((uint32_t)rem & 0xFFFFu) << 16);   // tensor_dim1 hi=0 | tile_dim0 = rem
        g1[4] = (int)NCH;                                 // tile_dim1 = 85, tile_dim2 = 0
        g1[5] = (int)(uint32_t)HW;                        // tensor_dim0_stride[31:0]
        g1[6] = 0;                                        // stride0 hi | stride1 lo (unused, 2D)
        g1[7] = 0;

        int4v gz4  = {0, 0, 0, 0};                        // groups 2/3: NULL (2-D tensor)
        int8v gz8  = {0, 0, 0, 0, 0, 0, 0, 0};            // extra group (clang-23 6-arg form)
        __builtin_amdgcn_tensor_load_to_lds(g0, g1, gz4, gz4, gz8, 0);
        __builtin_amdgcn_s_wait_tensorcnt(0);
    }
    __syncthreads();

    // ---- transform + fully coalesced channel-minor writeback ----
    const float aw = (a == 0) ? a0w : (a == 1) ? a1w : a2w;
    const float ah = (a == 0) ? a0h : (a == 1) ? a1h : a2h;
    const int rowBase = b * 22743 + scaleRowOff + a * HW + s0;  // output box row
    const int total   = NCH * rem;

    for (int j = (int)threadIdx.x; j < total; j += 256) {
        int s = j / NCH;            // spatial within tile
        int c = j - s * NCH;        // channel 0..84
        float v = lds[c * rowStride + s];
        float r;
        if (c >= 4) {
            r = 1.0f / (1.0f + __expf(-v));                 // conf + class sigmoid
        } else if (c == 2) {
            r = __expf(v) * aw;                             // bw
        } else if (c == 3) {
            r = __expf(v) * ah;                             // bh
        } else {
            int spat = s0 + s;
            int gy = spat / W;
            int gx = spat - gy * W;
            float g = (c == 0) ? (float)gx : (float)gy;
            r = (1.0f / (1.0f + __expf(-v)) + g) * strideF; // bx / by
        }
        // j = s*85 + c, and output rows are contiguous -> coalesced
        out[(long long)rowBase * NCH + j] = r;
    }
}

extern "C" void kernel_launch(void* const* d_in, const int* in_sizes, int n_in,
                              void* d_out, int out_size, void* d_ws, size_t ws_size,
                              hipStream_t stream) {
    (void)in_sizes; (void)n_in; (void)out_size; (void)d_ws; (void)ws_size;
    const float* x1 = (const float*)d_in[0];   // [16,255,76,76]
    const float* x2 = (const float*)d_in[1];   // [16,255,38,38]
    const float* x3 = (const float*)d_in[2];   // [16,255,19,19]
    float* out = (float*)d_out;                // [16,22743,85]

    // scale 1: stride 8, anchors (10,13)(16,30)(33,23), rows [0, 17328)
    {
        int W = 76, HW = 5776;
        dim3 grid((HW + TILE_S - 1) / TILE_S, 3, 16);
        yolo_decode_tdm<<<grid, dim3(256), 0, stream>>>(
            x1, out, W, HW, 8.0f, 0,
            10.f, 13.f, 16.f, 30.f, 33.f, 23.f);
    }
    // scale 2: stride 16, anchors (30,61)(62,45)(59,119), rows [17328, 21660)
    {
        int W = 38, HW = 1444;
        dim3 grid((HW + TILE_S - 1) / TILE_S, 3, 16);
        yolo_decode_tdm<<<grid, dim3(256), 0, stream>>>(
            x2, out, W, HW, 16.0f, 3 * 5776,
            30.f, 61.f, 62.f, 45.f, 59.f, 119.f);
    }
    // scale 3: stride 32, anchors (116,90)(156,198)(373,326), rows [21660, 22743)
    {
        int W = 19, HW = 361;
        dim3 grid((HW + TILE_S - 1) / TILE_S, 3, 16);
        yolo_decode_tdm<<<grid, dim3(256), 0, stream>>>(
            x3, out, W, HW, 32.0f, 3 * 5776 + 3 * 1444,
            116.f, 90.f, 156.f, 198.f, 373.f, 326.f);
    }
}